// DynamicConvolutionAttention_20375324852831
// MI455X (gfx1250) — compile-verified
//
#include <hip/hip_runtime.h>
#include <hip/hip_bf16.h>

// Dynamic Convolution Attention for MI455X (gfx1250, wave32).
// Pipeline: [A] dynamic filters G  [B] log-prior  [D] fused f/g conv + energy
// MLP via V_WMMA_F32_16X16X4_F32  [E] softmax  [F] context reduction (B128).

#define B_    64
#define T_    1024
#define QD    1024
#define MD    512
#define AD    128
#define NS    8
#define SK    21
#define ND    8
#define DK    21
#define PL    11
#define NFEAT 16
#define GSZ   (ND * DK)   // 168

typedef __attribute__((ext_vector_type(2))) float v2f;
typedef __attribute__((ext_vector_type(8))) float v8f;

// Branch-free tanh: sign(x) * (1 - e) / (1 + e), e = exp(-2|x|).
// exp(-2|x|) = exp2(-2*log2(e)*|x|); uses v_exp_f32 / v_rcp_f32, no branches.
__device__ __forceinline__ float fast_tanh(float x) {
  const float ax = __builtin_fabsf(x);
  const float e  = __builtin_amdgcn_exp2f(-2.885390081777927f * ax);
  const float r  = (1.f - e) * __builtin_amdgcn_rcpf(1.f + e);
  return __builtin_copysignf(r, x);
}

// ---------------------------------------------------------------- kernel A
// h = tanh(W_w @ q[b] + W_b);  G[b] = V_w @ h
__global__ __launch_bounds__(128) void dca_dyn_filters(
    const float* __restrict__ query, const float* __restrict__ W_w,
    const float* __restrict__ W_b, const float* __restrict__ V_w,
    float* __restrict__ G) {
  __shared__ float q_s[QD];
  __shared__ float h_s[AD];
  const int b = blockIdx.x, tid = threadIdx.x;
  for (int i = tid; i < QD; i += 128) q_s[i] = query[b * QD + i];
  __syncthreads();
  float acc = W_b[tid];
  const float* wr = W_w + (size_t)tid * QD;
  for (int k = 0; k < QD; ++k) acc += wr[k] * q_s[k];
  h_s[tid] = fast_tanh(acc);
  __syncthreads();
  for (int m = tid; m < GSZ; m += 128) {
    float a = 0.f;
    const float* vr = V_w + m * AD;
    for (int k = 0; k < AD; ++k) a += vr[k] * h_s[k];
    G[b * GSZ + m] = a;
  }
}

// ---------------------------------------------------------------- kernel B
// p[b,t] = log(clip(causal_conv(aw, prior), 1e-6))
__global__ __launch_bounds__(256) void dca_prior(
    const float* __restrict__ aw, const float* __restrict__ prior,
    float* __restrict__ p) {
  const int idx = blockIdx.x * 256 + threadIdx.x;
  const int b = idx >> 10, t = idx & (T_ - 1);
  float s = 0.f;
#pragma unroll
  for (int k = 0; k < PL; ++k) {
    const int src = t + k - (PL - 1);
    if (src >= 0) s += aw[b * T_ + src] * prior[k];
  }
  // log(x) = log2(x) * ln2 via hardware v_log_f32
  p[idx] = __builtin_amdgcn_logf(fmaxf(s, 1e-6f)) * 0.6931471805599453f;
}

// ---------------------------------------------------------------- kernel D
// Per wave: 16 timesteps. Build A = X[16x16] (8 static + 8 dynamic 21-tap
// convs) in WMMA A-fragment layout, multiply by M = [U_w | T_w] (128x16)
// with V_WMMA_F32_16X16X4_F32, tanh + dot(v_w) + prior -> energy[b,t].
__global__ __launch_bounds__(512) void dca_energy(
    const float* __restrict__ aw, const float* __restrict__ F_w,
    const float* __restrict__ Gdyn, const float* __restrict__ U_w,
    const float* __restrict__ T_w, const float* __restrict__ T_b,
    const float* __restrict__ v_w, const float* __restrict__ p,
    float* __restrict__ energy) {
  __shared__ float s_aw[256 + SK - 1];  // aw chunk + halo (zero padded)
  __shared__ float s_filt[NFEAT * SK];  // rows 0-7 static, 8-15 dynamic
  __shared__ float s_M[AD * NFEAT];     // M[j][i] = i<8 ? U_w[j][i] : T_w[j][i-8]
  __shared__ float s_Tb[AD];
  __shared__ float s_vw[AD];

  const int b = blockIdx.x >> 2;        // T_/256 = 4 chunks per batch
  const int chunk = blockIdx.x & 3;
  const int tid = threadIdx.x;

  for (int i = tid; i < 256 + SK - 1; i += 512) {
    const int g = chunk * 256 - (SK / 2) + i;
    s_aw[i] = (g >= 0 && g < T_) ? aw[b * T_ + g] : 0.f;
  }
  for (int i = tid; i < NFEAT * SK; i += 512) {
    const int j = i / SK, k = i - j * SK;
    s_filt[i] = (j < NS) ? F_w[j * SK + k] : Gdyn[b * GSZ + (j - NS) * DK + k];
  }
  for (int i = tid; i < AD * NFEAT; i += 512) {
    const int j = i >> 4, ii = i & 15;
    s_M[i] = (ii < NS) ? U_w[j * NS + ii] : T_w[j * ND + (ii - NS)];
  }
  if (tid < AD) { s_Tb[tid] = T_b[tid]; s_vw[tid] = v_w[tid]; }
  __syncthreads();

  const int wave = tid >> 5, lane = tid & 31;
  const int col = lane & 15;            // A row (time) / B column (hidden)
  const bool hi = lane >= 16;           // K split: lo half K{0,1}, hi K{2,3}

  // A fragments: afrag[kk] covers K = 4*kk .. 4*kk+3 of the 16x16 X tile.
  v2f afrag[4];
#pragma unroll
  for (int kk = 0; kk < 4; ++kk) { afrag[kk].x = 0.f; afrag[kk].y = 0.f; }
  const int awbase = wave * 16 + col;
#pragma unroll
  for (int k = 0; k < SK; ++k) {
    const float a = s_aw[awbase + k];
#pragma unroll
    for (int kk = 0; kk < 4; ++kk) {
      const int fb = 4 * kk + (hi ? 2 : 0);
      afrag[kk].x += a * s_filt[fb * SK + k];
      afrag[kk].y += a * s_filt[(fb + 1) * SK + k];
    }
  }

  float partial[8];
#pragma unroll
  for (int r = 0; r < 8; ++r) partial[r] = 0.f;

  for (int nt = 0; nt < 8; ++nt) {       // 8 N-tiles cover 128 hidden units
    v8f c = {0.f, 0.f, 0.f, 0.f, 0.f, 0.f, 0.f, 0.f};
#pragma unroll
    for (int kk = 0; kk < 4; ++kk) {
      const int j = 16 * nt + col;
      const int ii = 4 * kk + (hi ? 2 : 0);
      v2f bfrag;
      bfrag.x = s_M[j * NFEAT + ii];
      bfrag.y = s_M[j * NFEAT + ii + 1];
      c = __builtin_amdgcn_wmma_f32_16x16x4_f32(
          false, afrag[kk], false, bfrag, (short)0, c, false, false);
    }
    const float tb = s_Tb[16 * nt + col];
    const float vw = s_vw[16 * nt + col];
#pragma unroll
    for (int r = 0; r < 8; ++r) partial[r] += vw * fast_tanh(c[r] + tb);
  }

  // Reduce across the 16 lanes of each half-wave (xor 1,2,4,8 stay in-half).
#pragma unroll
  for (int r = 0; r < 8; ++r) {
#pragma unroll
    for (int m = 8; m >= 1; m >>= 1)
      partial[r] += __shfl_xor(partial[r], m, 32);
  }
  const int t0 = chunk * 256 + wave * 16;
#pragma unroll
  for (int r = 0; r < 8; ++r) {
    if (col == r) {                      // lanes 0-7 write m=r; 16-23 m=r+8
      const int t = t0 + r + (hi ? 8 : 0);
      energy[b * T_ + t] = partial[r] + p[b * T_ + t];
    }
  }
}

// ---------------------------------------------------------------- kernel E
__global__ __launch_bounds__(256) void dca_softmax(float* __restrict__ e) {
  __shared__ float red[256];
  const int b = blockIdx.x, tid = threadIdx.x;
  float* row = e + b * T_;
  float v[4];
  float lm = -3.0e38f;
#pragma unroll
  for (int i = 0; i < 4; ++i) { v[i] = row[tid + i * 256]; lm = fmaxf(lm, v[i]); }
  red[tid] = lm; __syncthreads();
  for (int s = 128; s > 0; s >>= 1) {
    if (tid < s) red[tid] = fmaxf(red[tid], red[tid + s]);
    __syncthreads();
  }
  const float mx = red[0]; __syncthreads();
  float ls = 0.f;
#pragma unroll
  for (int i = 0; i < 4; ++i) {
    // exp(x) = exp2(x * log2 e) via hardware v_exp_f32
    v[i] = __builtin_amdgcn_exp2f((v[i] - mx) * 1.4426950408889634f);
    ls += v[i];
  }
  red[tid] = ls; __syncthreads();
  for (int s = 128; s > 0; s >>= 1) {
    if (tid < s) red[tid] += red[tid + s];
    __syncthreads();
  }
  const float inv = 1.f / red[0];
#pragma unroll
  for (int i = 0; i < 4; ++i) row[tid + i * 256] = v[i] * inv;
}

// ---------------------------------------------------------------- kernel F
__global__ __launch_bounds__(256) void dca_zero(float* __restrict__ out, int n) {
  const int i = blockIdx.x * 256 + threadIdx.x;
  if (i < n) out[i] = 0.f;
}

// context[b,:] += sum_t attw[b,t] * memory[b,t,:]  (float4 / B128 streams)
__global__ __launch_bounds__(128) void dca_context(
    const float* __restrict__ attw, const float* __restrict__ memory,
    float* __restrict__ out) {
  __shared__ float s_a[128];
  const int b = blockIdx.x >> 3, ch = blockIdx.x & 7, tid = threadIdx.x;
  s_a[tid] = attw[b * T_ + ch * 128 + tid];
  __syncthreads();
  float4 acc = make_float4(0.f, 0.f, 0.f, 0.f);
  const float4* mb =
      (const float4*)(memory + ((size_t)b * T_ + (size_t)ch * 128) * MD);
  for (int t = 0; t < 128; ++t) {
    const float a = s_a[t];
    const float4 m = mb[(size_t)t * (MD / 4) + tid];
    acc.x += a * m.x;
    acc.y += a * m.y;
    acc.z += a * m.z;
    acc.w += a * m.w;
  }
  float* o = out + b * MD + 4 * tid;
  atomicAdd(o + 0, acc.x);
  atomicAdd(o + 1, acc.y);
  atomicAdd(o + 2, acc.z);
  atomicAdd(o + 3, acc.w);
}

// ---------------------------------------------------------------- launch
extern "C" void kernel_launch(void* const* d_in, const int* in_sizes, int n_in,
                              void* d_out, int out_size, void* d_ws,
                              size_t ws_size, hipStream_t stream) {
  const float* query  = (const float*)d_in[0];
  const float* memory = (const float*)d_in[1];
  const float* aw     = (const float*)d_in[2];
  const float* W_w    = (const float*)d_in[3];
  const float* W_b    = (const float*)d_in[4];
  const float* V_w    = (const float*)d_in[5];
  const float* F_w    = (const float*)d_in[6];
  const float* U_w    = (const float*)d_in[7];
  const float* T_w    = (const float*)d_in[8];
  const float* T_b    = (const float*)d_in[9];
  const float* v_w    = (const float*)d_in[10];
  const float* prior  = (const float*)d_in[11];
  float* out = (float*)d_out;

  float* ws     = (float*)d_ws;
  float* G      = ws;                  // B_*GSZ  = 10752 floats
  float* p      = G + B_ * GSZ;        // B_*T_   = 65536 floats
  float* energy = p + B_ * T_;         // B_*T_   = 65536 floats (attw in-place)

  dca_dyn_filters<<<B_, 128, 0, stream>>>(query, W_w, W_b, V_w, G);
  dca_prior<<<(B_ * T_) / 256, 256, 0, stream>>>(aw, prior, p);
  dca_energy<<<B_ * 4, 512, 0, stream>>>(aw, F_w, G, U_w, T_w, T_b, v_w, p,
                                         energy);
  dca_softmax<<<B_, 256, 0, stream>>>(energy);
  dca_zero<<<(B_ * MD + 255) / 256, 256, 0, stream>>>(out, B_ * MD);
  dca_context<<<B_ * 8, 128, 0, stream>>>(energy, memory, out);
}